// SingleGAT_17136919511420
// MI455X (gfx1250) — compile-verified
//
#include <hip/hip_runtime.h>
#include <math.h>

#define NEG_SLOPE 0.2f

typedef __attribute__((ext_vector_type(2))) float v2f;
typedef __attribute__((ext_vector_type(8))) float v8f;

// ---------- atomics ----------
// Native f32 add (agent scope, relaxed) -> global_atomic_add_f32
__device__ __forceinline__ void atomic_add_f32(float* p, float v) {
  __hip_atomic_fetch_add(p, v, __ATOMIC_RELAXED, __HIP_MEMORY_SCOPE_AGENT);
}

// Float max via sign-split integer atomics (monotone bit orderings):
// sign clear  -> signed  atomic max
// sign set    -> unsigned atomic min
// init with 0xff800000 (-inf). Correct across mixed signs (incl. -0.0).
__device__ __forceinline__ void atomic_max_f32(float* p, float v) {
  int bits = __float_as_int(v);
  if (bits >= 0) {
    __hip_atomic_fetch_max((int*)p, bits, __ATOMIC_RELAXED, __HIP_MEMORY_SCOPE_AGENT);
  } else {
    __hip_atomic_fetch_min((unsigned int*)p, (unsigned int)bits,
                           __ATOMIC_RELAXED, __HIP_MEMORY_SCOPE_AGENT);
  }
}

__device__ __forceinline__ float leaky(float e) {
  return e > 0.0f ? e : NEG_SLOPE * e;
}

// edge i in [0, E+N): first E from edge_index (int64), last N are self loops
__device__ __forceinline__ void edge_sd(const long long* __restrict__ ei,
                                        long long E, long long i, int* s, int* d) {
  if (i < E) { *s = (int)ei[i]; *d = (int)ei[E + i]; }
  else       { *s = *d = (int)(i - E); }
}

// ---------- init scratch ----------
__global__ void k_init(float* out1acc, float* m1, float* dn1,
                       float* m2, float* dn2, float* out2, int N) {
  long long i = (long long)blockIdx.x * blockDim.x + threadIdx.x;
  if (i < (long long)N * 64) out1acc[i] = 0.0f;
  if (i < (long long)N * 2) { m1[i] = __int_as_float(0xff800000); dn1[i] = 0.0f; }
  if (i < N) { m2[i] = __int_as_float(0xff800000); dn2[i] = 0.0f; out2[i] = 0.0f; }
}

// ---------- layer-1 feature transform: h1[N,64] = x[N,4] @ W1[4,64] ----------
// One wave -> one 16-node row tile; 4 column tiles of 16 via V_WMMA_F32_16X16X4_F32.
// A (16x4 f32, 2 VGPR): lanes 0-15 hold K=0/1 for M=lane; lanes 16-31 hold K=2/3.
// B (4x16 f32, 2 VGPR): same K split, N = lane&15.
// D (16x16 f32, 8 VGPR): VGPR r -> row r (lanes 0-15) / row r+8 (lanes 16-31), col = lane&15.
__global__ void k_gemm1(const float* __restrict__ x, const float* __restrict__ W1,
                        float* __restrict__ h1, int N) {
  int lane = threadIdx.x & 31;
  int wave = (int)(((long long)blockIdx.x * blockDim.x + threadIdx.x) >> 5);
  int numTiles = (N + 15) >> 4;
  if (wave >= numTiles) return;   // wave-uniform: EXEC stays all-ones for WMMA

  int m  = lane & 15;
  int kb = (lane >> 4) << 1;      // 0 (lanes 0-15) or 2 (lanes 16-31)
  int node = wave * 16 + m;
  int nc = node < N ? node : N - 1;
  // wave-uniform: tile fully in-bounds? (always true when 16 | N)
  bool full = (wave * 16 + 16) <= N;
  int rowbase = wave * 16 + ((lane < 16) ? 0 : 8);

#if __has_builtin(__builtin_amdgcn_wmma_f32_16x16x4_f32)
  v2f a;
  a.x = x[nc * 4 + kb + 0];
  a.y = x[nc * 4 + kb + 1];
#pragma unroll
  for (int ct = 0; ct < 4; ++ct) {
    int col = ct * 16 + m;
    v2f b;
    b.x = W1[(kb + 0) * 64 + col];
    b.y = W1[(kb + 1) * 64 + col];
    v8f c = {};
    c = __builtin_amdgcn_wmma_f32_16x16x4_f32(false, a, false, b,
                                              (short)0, c, false, false);
    float* p = &h1[(long long)rowbase * 64 + col];
    if (full) {
      // fast path: 8 plain strided b32 stores, no EXEC manipulation
#pragma unroll
      for (int r = 0; r < 8; ++r) p[(long long)r * 64] = c[r];
    } else {
#pragma unroll
      for (int r = 0; r < 8; ++r)
        if (rowbase + r < N) p[(long long)r * 64] = c[r];
    }
  }
#else
  if (lane < 16 && node < N) {
    for (int c0 = 0; c0 < 64; ++c0) {
      float acc = 0.0f;
      for (int k = 0; k < 4; ++k) acc += x[nc * 4 + k] * W1[k * 64 + c0];
      h1[(long long)node * 64 + c0] = acc;
    }
  }
  (void)full; (void)rowbase;
#endif
}

// ---------- attention logits: as1/ad1[N,2] ----------
__global__ void k_alpha1(const float* __restrict__ h1,
                         const float* __restrict__ a_src, const float* __restrict__ a_dst,
                         float* __restrict__ as1, float* __restrict__ ad1, int N) {
  int n = blockIdx.x * blockDim.x + threadIdx.x;
  if (n >= N) return;
#pragma unroll
  for (int hh = 0; hh < 2; ++hh) {
    float ss = 0.0f, dd = 0.0f;
#pragma unroll 8
    for (int c = 0; c < 32; ++c) {
      float v = h1[(long long)n * 64 + hh * 32 + c];
      ss += v * a_src[hh * 32 + c];
      dd += v * a_dst[hh * 32 + c];
    }
    as1[n * 2 + hh] = ss;
    ad1[n * 2 + hh] = dd;
  }
}

// ---------- layer-1 edge pass 1: segment max ----------
__global__ void k_emax1(const long long* __restrict__ ei, long long E, long long ET,
                        const float* __restrict__ as1, const float* __restrict__ ad1,
                        float* __restrict__ m1) {
  long long i = (long long)blockIdx.x * blockDim.x + threadIdx.x;
  if (i >= ET) return;
  int s, d; edge_sd(ei, E, i, &s, &d);
#pragma unroll
  for (int hh = 0; hh < 2; ++hh) {
    float e = leaky(as1[s * 2 + hh] + ad1[d * 2 + hh]);
    atomic_max_f32(&m1[d * 2 + hh], e);
  }
}

// ---------- layer-1 edge pass 2: segment sum of exp ----------
__global__ void k_esum1(const long long* __restrict__ ei, long long E, long long ET,
                        const float* __restrict__ as1, const float* __restrict__ ad1,
                        const float* __restrict__ m1, float* __restrict__ dn1) {
  long long i = (long long)blockIdx.x * blockDim.x + threadIdx.x;
  if (i >= ET) return;
  int s, d; edge_sd(ei, E, i, &s, &d);
#pragma unroll
  for (int hh = 0; hh < 2; ++hh) {
    float e = leaky(as1[s * 2 + hh] + ad1[d * 2 + hh]);
    float ex = __expf(e - m1[d * 2 + hh]);
    atomic_add_f32(&dn1[d * 2 + hh], ex);
  }
}

// ---------- layer-1 edge pass 3: weighted message scatter (16 lanes / edge) ----------
__global__ void k_emsg1(const long long* __restrict__ ei, long long E, long long ET,
                        const float* __restrict__ as1, const float* __restrict__ ad1,
                        const float* __restrict__ m1, const float* __restrict__ dn1,
                        const float* __restrict__ h1, float* __restrict__ out1acc) {
  long long t = (long long)blockIdx.x * blockDim.x + threadIdx.x;
  long long i = t >> 4;
  int part = (int)(t & 15);        // 16 lanes per edge, 4 floats each
  if (i >= ET) return;
  // prefetch the streaming edge-index rows a few KB ahead (global_prefetch_b8)
  long long ip = i + 4096;
  if (ip < E) {
    __builtin_prefetch(&ei[ip], 0, 1);
    __builtin_prefetch(&ei[E + ip], 0, 1);
  }
  int s, d; edge_sd(ei, E, i, &s, &d);
  int hh = part >> 3;              // head 0: parts 0-7, head 1: parts 8-15
  float e = leaky(as1[s * 2 + hh] + ad1[d * 2 + hh]);
  float ex = __expf(e - m1[d * 2 + hh]);
  float alpha = ex / (dn1[d * 2 + hh] + 1e-16f);
  const float4 hv = *(const float4*)&h1[(long long)s * 64 + part * 4];
  float* o = &out1acc[(long long)d * 64 + part * 4];
  atomic_add_f32(o + 0, hv.x * alpha);
  atomic_add_f32(o + 1, hv.y * alpha);
  atomic_add_f32(o + 2, hv.z * alpha);
  atomic_add_f32(o + 3, hv.w * alpha);
}

// ---------- between layers: bias+relu, z = h @ W2, layer-2 logits ----------
__global__ void k_mid(const float* __restrict__ out1acc, const float* __restrict__ b1,
                      const float* __restrict__ W2,
                      const float* __restrict__ a_src2, const float* __restrict__ a_dst2,
                      float* __restrict__ z, float* __restrict__ as2, float* __restrict__ ad2,
                      int N) {
  int n = blockIdx.x * blockDim.x + threadIdx.x;
  if (n >= N) return;
  float acc = 0.0f;
#pragma unroll 8
  for (int k = 0; k < 64; ++k) {
    float v = out1acc[(long long)n * 64 + k] + b1[k];
    v = v > 0.0f ? v : 0.0f;     // relu between layers
    acc += v * W2[k];
  }
  z[n] = acc;
  as2[n] = acc * a_src2[0];
  ad2[n] = acc * a_dst2[0];
}

// ---------- layer-2 edge passes (H=1, C=1) ----------
__global__ void k_emax2(const long long* __restrict__ ei, long long E, long long ET,
                        const float* __restrict__ as2, const float* __restrict__ ad2,
                        float* __restrict__ m2) {
  long long i = (long long)blockIdx.x * blockDim.x + threadIdx.x;
  if (i >= ET) return;
  int s, d; edge_sd(ei, E, i, &s, &d);
  atomic_max_f32(&m2[d], leaky(as2[s] + ad2[d]));
}

__global__ void k_esum2(const long long* __restrict__ ei, long long E, long long ET,
                        const float* __restrict__ as2, const float* __restrict__ ad2,
                        const float* __restrict__ m2, float* __restrict__ dn2) {
  long long i = (long long)blockIdx.x * blockDim.x + threadIdx.x;
  if (i >= ET) return;
  int s, d; edge_sd(ei, E, i, &s, &d);
  float ex = __expf(leaky(as2[s] + ad2[d]) - m2[d]);
  atomic_add_f32(&dn2[d], ex);
}

__global__ void k_emsg2(const long long* __restrict__ ei, long long E, long long ET,
                        const float* __restrict__ as2, const float* __restrict__ ad2,
                        const float* __restrict__ m2, const float* __restrict__ dn2,
                        const float* __restrict__ z, float* __restrict__ out2) {
  long long i = (long long)blockIdx.x * blockDim.x + threadIdx.x;
  if (i >= ET) return;
  int s, d; edge_sd(ei, E, i, &s, &d);
  float ex = __expf(leaky(as2[s] + ad2[d]) - m2[d]);
  float alpha = ex / (dn2[d] + 1e-16f);
  atomic_add_f32(&out2[d], z[s] * alpha);
}

// ---------- final: sigmoid(out2 + b2) ----------
__global__ void k_final(const float* __restrict__ out2, const float* __restrict__ b2,
                        float* __restrict__ out, int N) {
  int n = blockIdx.x * blockDim.x + threadIdx.x;
  if (n >= N) return;
  float v = out2[n] + b2[0];
  out[n] = 1.0f / (1.0f + __expf(-v));
}

// ---------- launcher ----------
extern "C" void kernel_launch(void* const* d_in, const int* in_sizes, int n_in,
                              void* d_out, int out_size, void* d_ws, size_t ws_size,
                              hipStream_t stream) {
  const float*     x   = (const float*)d_in[0];       // [N,4]
  const long long* ei  = (const long long*)d_in[1];   // [2,E] int64
  const float*     W1  = (const float*)d_in[2];       // [4,64]
  const float*     aS1 = (const float*)d_in[3];       // [2,32]
  const float*     aD1 = (const float*)d_in[4];       // [2,32]
  const float*     b1  = (const float*)d_in[5];       // [64]
  const float*     W2  = (const float*)d_in[6];       // [64,1]
  const float*     aS2 = (const float*)d_in[7];       // [1,1]
  const float*     aD2 = (const float*)d_in[8];       // [1,1]
  const float*     b2  = (const float*)d_in[9];       // [1]
  float*           out = (float*)d_out;               // [N,1]

  const int N = in_sizes[0] / 4;
  const long long E  = (long long)(in_sizes[1] / 2);
  const long long ET = E + N;                         // with self loops

  // scratch carve (floats), ~142*N*4 bytes total (~28 MB for N=50000)
  float* w = (float*)d_ws;
  float* h1      = w; w += (size_t)N * 64;
  float* out1acc = w; w += (size_t)N * 64;
  float* as1     = w; w += (size_t)N * 2;
  float* ad1     = w; w += (size_t)N * 2;
  float* m1      = w; w += (size_t)N * 2;
  float* dn1     = w; w += (size_t)N * 2;
  float* z       = w; w += (size_t)N;
  float* as2     = w; w += (size_t)N;
  float* ad2     = w; w += (size_t)N;
  float* m2      = w; w += (size_t)N;
  float* dn2     = w; w += (size_t)N;
  float* out2    = w; w += (size_t)N;

  const int B = 256;
  long long initN = (long long)N * 64;
  int gInit  = (int)((initN + B - 1) / B);
  int gNode  = (N + B - 1) / B;
  int gTiles = (int)((((long long)((N + 15) / 16)) * 32 + B - 1) / B);  // 1 wave / tile
  int gEdge  = (int)((ET + B - 1) / B);
  int gMsg1  = (int)((ET * 16 + B - 1) / B);

  k_init  <<<gInit,  B, 0, stream>>>(out1acc, m1, dn1, m2, dn2, out2, N);
  k_gemm1 <<<gTiles, B, 0, stream>>>(x, W1, h1, N);
  k_alpha1<<<gNode,  B, 0, stream>>>(h1, aS1, aD1, as1, ad1, N);
  k_emax1 <<<gEdge,  B, 0, stream>>>(ei, E, ET, as1, ad1, m1);
  k_esum1 <<<gEdge,  B, 0, stream>>>(ei, E, ET, as1, ad1, m1, dn1);
  k_emsg1 <<<gMsg1,  B, 0, stream>>>(ei, E, ET, as1, ad1, m1, dn1, h1, out1acc);
  k_mid   <<<gNode,  B, 0, stream>>>(out1acc, b1, W2, aS2, aD2, z, as2, ad2, N);
  k_emax2 <<<gEdge,  B, 0, stream>>>(ei, E, ET, as2, ad2, m2);
  k_esum2 <<<gEdge,  B, 0, stream>>>(ei, E, ET, as2, ad2, m2, dn2);
  k_emsg2 <<<gEdge,  B, 0, stream>>>(ei, E, ET, as2, ad2, m2, dn2, z, out2);
  k_final <<<gNode,  B, 0, stream>>>(out2, b2, out, N);
}